// NonLocalBlockND_39487929319556
// MI455X (gfx1250) — compile-verified
//
#include <hip/hip_runtime.h>
#include <hip/hip_bf16.h>

// Problem constants (from reference): N=1024, T=300, M=2, V=25, C=3
#define NB        1024          // batch n
#define TDIM      300
#define NCH       150           // M*V*C channels per n
#define BTOT      2048          // N*M
#define RES_ELEMS 46080000      // T * B * V * C
#define DIS_OFF   46080000
#define CEN_OFF   46131200      // DIS_OFF + B*V
#define NTRIPLES  15360000      // RES_ELEMS / 3

typedef __attribute__((ext_vector_type(2))) float v2f;
typedef __attribute__((ext_vector_type(8))) float v8f;

// ---------------------------------------------------------------------------
// Kernel 1: per-n temporal column sums via WMMA (A = ones(16x4) so
// D[i,j] = sum_k B[k,j] + C[i,j]), then tiny attention -> anchor.
// One block per n (1024 blocks, 256 threads = 8 waves).
// ---------------------------------------------------------------------------
__global__ __launch_bounds__(256) void nlb_reduce_anchor(
    const float* __restrict__ x,
    const float* __restrict__ g_w,  const float* __restrict__ g_b,
    const float* __restrict__ th_w, const float* __restrict__ th_b,
    const float* __restrict__ ph_w, const float* __restrict__ ph_b,
    float* __restrict__ dis_out,     // (B, 25)
    float* __restrict__ center_out)  // (B, 3) == anchor (read by kernel 2)
{
    __shared__ float s[160];         // column sums, k = 0..149

    const int n    = blockIdx.x;
    const float* xn = x + (size_t)n * (TDIM * NCH);
    const int tid  = threadIdx.x;
    const int wave = tid >> 5;
    const int lane = tid & 31;

    // 10 column tiles of 16 channels; waves 0..7 take tile=wave, waves 0,1
    // additionally take tiles 8,9. Condition is wave-uniform -> EXEC all-1s.
    for (int tile = wave; tile < 10; tile += 8) {
        const int k0     = tile * 16;
        const int grp    = lane >> 4;      // 0 | 1
        const int col    = k0 + (lane & 15);
        const bool valid = (col < NCH);
        const int colc   = valid ? col : (NCH - 1);

        v2f a; a.x = 1.0f; a.y = 1.0f;     // A = ones(16x4)
        v8f c = {};

        #pragma unroll 5
        for (int tch = 0; tch < 75; ++tch) {
            const int t = tch * 4;
            // B (4x16 f32) layout: VGPR0 -> K = 2*grp, VGPR1 -> K = 2*grp+1
            float b0 = xn[(t + 2 * grp    ) * NCH + colc];
            float b1 = xn[(t + 2 * grp + 1) * NCH + colc];
            v2f b; b.x = valid ? b0 : 0.0f; b.y = valid ? b1 : 0.0f;
            c = __builtin_amdgcn_wmma_f32_16x16x4_f32(
                    /*neg_a=*/false, a, /*neg_b=*/false, b,
                    /*c_mod=*/(short)0, c, /*reuse_a=*/false, /*reuse_b=*/false);
        }
        // Row M=0 of D lives in c[0], lanes 0..15 (N = lane) -> column sums.
        if (lane < 16 && col < NCH) s[col] = c[0];
    }
    __syncthreads();

    // Tiny attention: 2 threads, one per person m (b = 2n + m).
    if (tid < 2) {
        const int m = tid;
        const int b = n * 2 + m;

        float xm[3][25];
        #pragma unroll
        for (int v = 0; v < 25; ++v)
            #pragma unroll
            for (int cc = 0; cc < 3; ++cc)
                xm[cc][v] = s[m * 75 + v * 3 + cc] * (1.0f / 300.0f);

        const float tw0 = th_w[0], tw1 = th_w[1], tw2 = th_w[2], tb = th_b[0];
        const float pw0 = ph_w[0], pw1 = ph_w[1], pw2 = ph_w[2], pb = ph_b[0];

        float theta[25];
        float phi_sum = 0.0f;
        #pragma unroll
        for (int v = 0; v < 25; ++v) {
            theta[v] = tb + tw0 * xm[0][v] + tw1 * xm[1][v] + tw2 * xm[2][v];
            phi_sum += pb + pw0 * xm[0][v] + pw1 * xm[1][v] + pw2 * xm[2][v];
        }

        // softmax(theta * phi_sum * 20) over V
        float mx = -INFINITY;
        float dr[25];
        #pragma unroll
        for (int v = 0; v < 25; ++v) {
            dr[v] = theta[v] * phi_sum * 20.0f;
            mx = fmaxf(mx, dr[v]);
        }
        float se = 0.0f;
        #pragma unroll
        for (int v = 0; v < 25; ++v) { dr[v] = __expf(dr[v] - mx); se += dr[v]; }
        const float inv = 1.0f / se;

        float anc0 = 0.0f, anc1 = 0.0f, anc2 = 0.0f;
        #pragma unroll
        for (int v = 0; v < 25; ++v) {
            const float d = dr[v] * inv;
            dis_out[b * 25 + v] = d;
            const float x0 = xm[0][v], x1 = xm[1][v], x2 = xm[2][v];
            anc0 += d * (g_b[0] + g_w[0] * x0 + g_w[1] * x1 + g_w[2] * x2);
            anc1 += d * (g_b[1] + g_w[3] * x0 + g_w[4] * x1 + g_w[5] * x2);
            anc2 += d * (g_b[2] + g_w[6] * x0 + g_w[7] * x1 + g_w[8] * x2);
        }
        center_out[b * 3 + 0] = anc0;
        center_out[b * 3 + 1] = anc1;
        center_out[b * 3 + 2] = anc2;
    }
}

// ---------------------------------------------------------------------------
// Kernel 2: res[t,b,v,:] = x[n,t,m*75+v*3 ..] - anchor[b,:]
// One xyz triple per lane; x read is the last use (hopefully hot in L2 from
// kernel 1), res written with non-temporal stores to avoid evicting x.
// ---------------------------------------------------------------------------
__global__ __launch_bounds__(256) void nlb_res(
    const float* __restrict__ x,
    const float* __restrict__ anchor,   // (B,3) = center_out region of d_out
    float* __restrict__ res)
{
    int idx = blockIdx.x * blockDim.x + threadIdx.x;
    const int stride = gridDim.x * blockDim.x;
    for (; idx < NTRIPLES; idx += stride) {
        // triple index -> (t, b, v)
        const int t = idx / 51200;          // 2048*25 triples per t
        const int r = idx - t * 51200;
        const int b = r / 25;
        const int v = r - b * 25;
        const int n = b >> 1;
        const int m = b & 1;
        // triple index into x: (n*45000 + t*150 + m*75 + v*3) / 3
        const int in3 = n * 15000 + t * 50 + m * 25 + v;

        const float* xp = x + (size_t)in3 * 3;
        const float x0 = __builtin_nontemporal_load(xp + 0);
        const float x1 = __builtin_nontemporal_load(xp + 1);
        const float x2 = __builtin_nontemporal_load(xp + 2);

        const float a0 = anchor[b * 3 + 0];
        const float a1 = anchor[b * 3 + 1];
        const float a2 = anchor[b * 3 + 2];

        float* op = res + (size_t)idx * 3;
        __builtin_nontemporal_store(x0 - a0, op + 0);
        __builtin_nontemporal_store(x1 - a1, op + 1);
        __builtin_nontemporal_store(x2 - a2, op + 2);
    }
}

extern "C" void kernel_launch(void* const* d_in, const int* in_sizes, int n_in,
                              void* d_out, int out_size, void* d_ws, size_t ws_size,
                              hipStream_t stream) {
    const float* x    = (const float*)d_in[0];
    const float* g_w  = (const float*)d_in[1];
    const float* g_b  = (const float*)d_in[2];
    const float* th_w = (const float*)d_in[3];
    const float* th_b = (const float*)d_in[4];
    const float* ph_w = (const float*)d_in[5];
    const float* ph_b = (const float*)d_in[6];

    float* out = (float*)d_out;
    float* res = out;                 // 46,080,000 floats
    float* dis = out + DIS_OFF;       // 51,200 floats
    float* cen = out + CEN_OFF;       // 6,144 floats (== anchor)

    nlb_reduce_anchor<<<NB, 256, 0, stream>>>(x, g_w, g_b, th_w, th_b,
                                              ph_w, ph_b, dis, cen);
    nlb_res<<<NTRIPLES / 256, 256, 0, stream>>>(x, cen, res);
}